// Encoder_17824114279155
// MI455X (gfx1250) — compile-verified
//
#include <hip/hip_runtime.h>

typedef __attribute__((ext_vector_type(2))) float v2f;
typedef __attribute__((ext_vector_type(8))) float v8f;

#define FEAT 64

// ---------------------------------------------------------------------------
// GEMM: H[N,64] = X[N,64] @ W[64,64] using V_WMMA_F32_16X16X4_F32.
// One wave computes one 16x16 output tile; K loop steps by 4 (16 WMMAs).
// A-layout (16x4 f32): lane L -> (M = L&15, K = 2*(L>>4)) , K+1 in .y
// B-layout (4x16 f32): lane L -> (K = 2*(L>>4), N = L&15) , K+1 in .y
// C/D-layout (16x16 f32): VGPR r -> M = r + 8*(lane>>4), N = lane&15
// ---------------------------------------------------------------------------
__global__ __launch_bounds__(256) void gemm64_wmma(const float* __restrict__ X,
                                                   const float* __restrict__ W,
                                                   float* __restrict__ H)
{
    const int tid    = threadIdx.x;
    const int wave   = tid >> 5;
    const int lane   = tid & 31;
    const int tileId = blockIdx.x * 8 + wave;   // 4 col-tiles per 16-row strip
    const int row0   = (tileId >> 2) * 16;
    const int col0   = (tileId & 3) * 16;

    const int m    = lane & 15;
    const int half = lane >> 4;     // 0 or 1
    const int kp   = half * 2;      // K sub-offset within a K=4 group

    v8f c = {0.f, 0.f, 0.f, 0.f, 0.f, 0.f, 0.f, 0.f};

    const float* xrow = X + (size_t)(row0 + m) * FEAT;

    #pragma unroll
    for (int k = 0; k < FEAT; k += 4) {
        v2f a, b;
        a.x = xrow[k + kp];
        a.y = xrow[k + kp + 1];
        b.x = W[(size_t)(k + kp)     * FEAT + col0 + m];
        b.y = W[(size_t)(k + kp + 1) * FEAT + col0 + m];
        c = __builtin_amdgcn_wmma_f32_16x16x4_f32(
                /*neg_a=*/false, a, /*neg_b=*/false, b,
                /*c_mod=*/(short)0, c, /*reuse_a=*/false, /*reuse_b=*/false);
    }

    float* out = H + (size_t)(row0 + half * 8) * FEAT + col0 + m;
    #pragma unroll
    for (int r = 0; r < 8; ++r)
        out[(size_t)r * FEAT] = c[r];
}

// ---------------------------------------------------------------------------
// Edge scatter-add: one wave per edge; each lane moves 2 consecutive floats.
// src/dst are wave-uniform -> scalar loads; gather is a float2 (b64) load;
// scatter is two global_atomic_add_f32 that resolve in L2 (working set is
// 25.6 MB, resident in the 192 MB L2).
// ---------------------------------------------------------------------------
__global__ __launch_bounds__(256) void scatter_add_edges(const float* __restrict__ H,
                                                         const int* __restrict__ src,
                                                         const int* __restrict__ dst,
                                                         float* __restrict__ AGG,
                                                         int nEdges)
{
    int gid  = blockIdx.x * blockDim.x + threadIdx.x;
    int edge = gid >> 5;
    if (edge >= nEdges) return;
    int lane = gid & 31;

    int s = src[edge];
    int d = dst[edge];

    const float2 v = *((const float2*)(H + (size_t)s * FEAT) + lane);
    float* ap = AGG + (size_t)d * FEAT + lane * 2;
    atomicAdd(ap,     v.x);
    atomicAdd(ap + 1, v.y);
}

// ---------------------------------------------------------------------------
// out[i] = relu(A[i] + bias[i & 63]), float4 vectorized, in-place capable.
// ---------------------------------------------------------------------------
__global__ __launch_bounds__(256) void bias_relu(float* __restrict__ A,
                                                 const float* __restrict__ bias,
                                                 int total)
{
    int idx = (blockIdx.x * blockDim.x + threadIdx.x) * 4;
    if (idx >= total) return;
    float4 v = *(float4*)(A + idx);
    int f = idx & (FEAT - 1);
    v.x = fmaxf(v.x + bias[f],     0.f);
    v.y = fmaxf(v.y + bias[f + 1], 0.f);
    v.z = fmaxf(v.z + bias[f + 2], 0.f);
    v.w = fmaxf(v.w + bias[f + 3], 0.f);
    *(float4*)(A + idx) = v;
}

extern "C" void kernel_launch(void* const* d_in, const int* in_sizes, int n_in,
                              void* d_out, int out_size, void* d_ws, size_t ws_size,
                              hipStream_t stream)
{
    const float* x   = (const float*)d_in[0];
    const int*   src = (const int*)  d_in[1];
    const int*   dst = (const int*)  d_in[2];
    const float* W1  = (const float*)d_in[3];
    const float* b1  = (const float*)d_in[4];
    const float* W2  = (const float*)d_in[5];
    const float* b2  = (const float*)d_in[6];
    float*       out = (float*)d_out;

    const int nNodes = in_sizes[0] / FEAT;   // 100000
    const int nEdges = in_sizes[1];          // 1280000
    const int total  = nNodes * FEAT;        // 6.4M floats
    const size_t featBytes = (size_t)total * sizeof(float);

    float* h   = (float*)d_ws;                            // 25.6 MB
    float* agg = (float*)((char*)d_ws + featBytes);       // 25.6 MB

    const int gemmBlocks    = (nNodes / 16) * 4 / 8;              // 3125
    const int scatterBlocks = (nEdges * 32 + 255) / 256;          // 160000
    const int ewBlocks      = (total / 4 + 255) / 256;            // 6250

    // ---- Layer 1: relu(segment_sum((x@W1)[src] -> dst) + b1) ----
    hipMemsetAsync(agg, 0, featBytes, stream);
    gemm64_wmma<<<gemmBlocks, 256, 0, stream>>>(x, W1, h);
    scatter_add_edges<<<scatterBlocks, 256, 0, stream>>>(h, src, dst, agg, nEdges);
    bias_relu<<<ewBlocks, 256, 0, stream>>>(agg, b1, total);

    // ---- Layer 2: relu(segment_sum((h1@W2)[src] -> dst) + b2) ----
    gemm64_wmma<<<gemmBlocks, 256, 0, stream>>>(agg, W2, h);
    hipMemsetAsync(out, 0, featBytes, stream);
    scatter_add_edges<<<scatterBlocks, 256, 0, stream>>>(h, src, dst, out, nEdges);
    bias_relu<<<ewBlocks, 256, 0, stream>>>(out, b2, total);
}